// CRF_90237262889186
// MI455X (gfx1250) — compile-verified
//
#include <hip/hip_runtime.h>
#include <hip/hip_bf16.h>

typedef __attribute__((ext_vector_type(2))) float v2f;
typedef __attribute__((ext_vector_type(8))) float v8f;

#define Bv 64
#define Tv 512
#define Dv 1024
#define Kv 32

__device__ __forceinline__ float fexp2(float x) { return __builtin_amdgcn_exp2f(x); }
__device__ __forceinline__ float flog2(float x) { return __builtin_amdgcn_logf(x); }

// Uniform broadcast of lane `l` (literal) via v_readlane_b32 -> SGPR operand.
__device__ __forceinline__ float readlane_f(float v, int l) {
    return __builtin_bit_cast(float,
        __builtin_amdgcn_readlane(__builtin_bit_cast(int, v), l));
}

// ---------------------------------------------------------------------------
// Kernel 0: transpose W [D,K] -> Wt [K,D] so WMMA B-fragments are contiguous.
// ---------------------------------------------------------------------------
__global__ __launch_bounds__(256) void crf_transposeW(const float* __restrict__ W,
                                                      float* __restrict__ Wt) {
    int idx = blockIdx.x * 256 + threadIdx.x;      // 0 .. 32767
    int k = idx >> 10;                             // 0..31
    int d = idx & 1023;                            // 0..1023
    Wt[idx] = W[d * Kv + k];                       // coalesced writes
}

// ---------------------------------------------------------------------------
// Kernel 1: emissions e[b,t,k] = x @ W + b  via V_WMMA_F32_16X16X4_F32.
// One wave per 16-row M-tile of the flattened (b*T+t) dimension.
// A frag (16x4 f32): lane&15 = M row, lane>>4 selects K pair {0,1}/{2,3}.
// B frag (4x16 f32): lane&15 = N col, lane>>4 selects K pair -> contiguous in Wt.
// Memory-bound: streams x (128 MB) at HBM rate; Wt (128 KB) stays L2-resident.
// ---------------------------------------------------------------------------
__global__ __launch_bounds__(128) void crf_emissions(const float* __restrict__ x,
                                                     const float* __restrict__ Wt,
                                                     const float* __restrict__ bias,
                                                     float* __restrict__ e) {
    const int lane = threadIdx.x & 31;
    const int wave = threadIdx.x >> 5;
    const int mt   = blockIdx.x * 4 + wave;        // 0 .. 2047 (32768/16 tiles)
    const int half = lane >> 4;                    // 0/1 -> K offset of 2
    const int l15  = lane & 15;

    const float* xptr = x  + (size_t)(mt * 16 + l15) * Dv + half * 2;
    const float* w0   = Wt + (size_t)(l15)      * Dv + half * 2;   // N tile 0
    const float* w1   = Wt + (size_t)(l15 + 16) * Dv + half * 2;   // N tile 1

    v8f acc0 = {};
    v8f acc1 = {};
    #pragma unroll 4
    for (int d = 0; d < Dv / 4; ++d) {
        v2f a  = *(const v2f*)(xptr + d * 4);
        v2f b0 = *(const v2f*)(w0   + d * 4);
        v2f b1 = *(const v2f*)(w1   + d * 4);
        acc0 = __builtin_amdgcn_wmma_f32_16x16x4_f32(false, a, false, b0,
                                                     (short)0, acc0, false, false);
        acc1 = __builtin_amdgcn_wmma_f32_16x16x4_f32(false, a, false, b1,
                                                     (short)0, acc1, false, false);
    }

    const float bv0 = bias[l15];
    const float bv1 = bias[l15 + 16];
    #pragma unroll
    for (int r = 0; r < 8; ++r) {
        // C/D layout: VGPR r -> M = r (lanes 0-15) / M = r+8 (lanes 16-31), N = l15
        size_t row = (size_t)mt * 16 + r + half * 8;
        e[row * Kv + l15]      = acc0[r] + bv0;
        e[row * Kv + l15 + 16] = acc1[r] + bv1;
    }
}

// ---------------------------------------------------------------------------
// Kernel 2: fused forward/backward recursions (scaled linear-space form).
//   LSE_j(a_j + U_jk) = m + log( sum_j exp(a_j - m) * expU[j,k] ),  m = max_j a_j
// exp(U) is hoisted out of the 512-step chain into 32 VGPRs per lane, so each
// step is: 5x shfl_xor max-reduce, 1x v_exp, 32x v_fmac with v_readlane
// scalar broadcast, 1x v_log. blocks 0..63 forward, 64..127 backward; one
// wave32 per chain, lane = tag index. beta is staged in d_out.
// ---------------------------------------------------------------------------
__global__ __launch_bounds__(32) void crf_scan(const float* __restrict__ e,
                                               const float* __restrict__ U,
                                               float* __restrict__ alpha_out,
                                               float* __restrict__ beta_out) {
    const int   lane   = threadIdx.x;
    const float LOG2E  = 1.44269504088896340736f;
    const float LN2    = 0.69314718055994530942f;
    float eU[Kv];

    if (blockIdx.x < Bv) {
        // -------- forward: alpha_t[k] = LSE_j(alpha[j] + U[j,k]) + e_t[k] -----
        const int bIdx = blockIdx.x;
        #pragma unroll
        for (int j = 0; j < Kv; ++j)
            eU[j] = fexp2(U[j * Kv + lane] * LOG2E);           // exp of column k=lane

        const float* eb = e         + (size_t)bIdx * Tv * Kv;
        float*       ab = alpha_out + (size_t)bIdx * Tv * Kv;

        float alpha = eb[lane];
        ab[lane] = alpha;
        for (int t = 1; t < Tv; ++t) {
            float m = alpha;
            #pragma unroll
            for (int off = 16; off; off >>= 1) m = fmaxf(m, __shfl_xor(m, off, 32));
            float p = fexp2((alpha - m) * LOG2E);              // lane j holds p_j
            float s = 0.f;
            #pragma unroll
            for (int j = 0; j < Kv; ++j)
                s = fmaf(readlane_f(p, j), eU[j], s);          // SGPR * VGPR fmac
            alpha = m + flog2(s) * LN2 + eb[t * Kv + lane];
            ab[t * Kv + lane] = alpha;
        }
    } else {
        // -------- backward: beta_t[j] = LSE_k(U[j,k] + e_{t+1}[k] + beta[k]) --
        const int bIdx = blockIdx.x - Bv;
        #pragma unroll
        for (int j = 0; j < Kv; ++j)
            eU[j] = fexp2(U[lane * Kv + j] * LOG2E);           // exp of row j=lane

        const float* eb = e        + (size_t)bIdx * Tv * Kv;
        float*       bb = beta_out + (size_t)bIdx * Tv * Kv;

        float beta = 0.f;
        bb[(Tv - 1) * Kv + lane] = 0.f;
        for (int t = Tv - 2; t >= 0; --t) {
            float g = eb[(t + 1) * Kv + lane] + beta;          // lane plays role k
            float m = g;
            #pragma unroll
            for (int off = 16; off; off >>= 1) m = fmaxf(m, __shfl_xor(m, off, 32));
            float q = fexp2((g - m) * LOG2E);                  // lane k holds q_k
            float s = 0.f;
            #pragma unroll
            for (int kk = 0; kk < Kv; ++kk)
                s = fmaf(readlane_f(q, kk), eU[kk], s);
            beta = m + flog2(s) * LN2;                         // lane plays role j
            bb[t * Kv + lane] = beta;
        }
    }
}

// ---------------------------------------------------------------------------
// Kernel 3: marg = softmax(alpha + beta) over k. beta is staged in d_out and
// overwritten in place. One wave per (b,t) row; shfl reductions.
// ---------------------------------------------------------------------------
__global__ __launch_bounds__(256) void crf_softmax(const float* __restrict__ alpha,
                                                   float* __restrict__ out) {
    const int lane = threadIdx.x & 31;
    const int wave = threadIdx.x >> 5;
    const size_t row = (size_t)blockIdx.x * 8 + wave;   // 0 .. 32767
    const size_t idx = row * Kv + lane;
    const float LOG2E = 1.44269504088896340736f;

    float s = alpha[idx] + out[idx];   // out currently holds beta
    float m = s;
    #pragma unroll
    for (int off = 16; off; off >>= 1) m = fmaxf(m, __shfl_xor(m, off, 32));
    float p = fexp2((s - m) * LOG2E);
    float sum = p;
    #pragma unroll
    for (int off = 16; off; off >>= 1) sum += __shfl_xor(sum, off, 32);
    out[idx] = p / sum;
}

// ---------------------------------------------------------------------------
extern "C" void kernel_launch(void* const* d_in, const int* in_sizes, int n_in,
                              void* d_out, int out_size, void* d_ws, size_t ws_size,
                              hipStream_t stream) {
    const float* x = (const float*)d_in[0];   // [64, 512, 1024]
    const float* W = (const float*)d_in[1];   // [1024, 32]
    const float* U = (const float*)d_in[2];   // [32, 32]
    const float* b = (const float*)d_in[3];   // [32]
    float* out = (float*)d_out;               // [64, 512, 32]

    float* ws    = (float*)d_ws;
    float* e     = ws;                           // 1,048,576 floats (4 MB)
    float* alpha = ws + (size_t)Bv * Tv * Kv;    // 1,048,576 floats (4 MB)
    float* Wt    = alpha + (size_t)Bv * Tv * Kv; // 32,768 floats (128 KB)

    crf_transposeW<<<dim3(128), dim3(256), 0, stream>>>(W, Wt);
    crf_emissions <<<dim3((Bv * Tv / 16) / 4), dim3(128), 0, stream>>>(x, Wt, b, e);
    crf_scan      <<<dim3(2 * Bv), dim3(32), 0, stream>>>(e, U, alpha, out);
    crf_softmax   <<<dim3(Bv * Tv / 8), dim3(256), 0, stream>>>(alpha, out);
}